// DTKSA_66425964200579
// MI455X (gfx1250) — compile-verified
//
#include <hip/hip_runtime.h>
#include <math.h>

// ---------------- problem constants ----------------
#define DIMC 384
#define HEADS 8
#define CH 48            // DIMC / HEADS
#define HSP 96
#define WSP 96
#define HWP (HSP*WSP)    // 9216 pooled spatial
#define HF 192
#define WF 192
#define HWF (HF*WF)      // 36864 full-res spatial
#define QKVC (3*DIMC)    // 1152

typedef __attribute__((ext_vector_type(2))) float v2f;
typedef __attribute__((ext_vector_type(8))) float v8f;
typedef __attribute__((ext_vector_type(4))) unsigned int u32x4;
typedef __attribute__((ext_vector_type(8))) int i32x8;
typedef __attribute__((ext_vector_type(4))) int i32x4;

// V_WMMA_F32_16X16X4_F32 : D(16x16,f32) = A(16x4,f32) * B(4x16,f32) + C
__device__ __forceinline__ v8f wmma4(v2f a, v2f b, v8f c) {
  return __builtin_amdgcn_wmma_f32_16x16x4_f32(false, a, false, b, (short)0, c,
                                               false, false);
}

__device__ __forceinline__ v8f v8zero() {
  v8f z = {0.f, 0.f, 0.f, 0.f, 0.f, 0.f, 0.f, 0.f};
  return z;
}

// ------------------------------------------------------------
// TDM: build D# descriptor (CDNA5 ISA)
// and issue TENSOR_LOAD_TO_LDS for a (tile_d0 x tile_d1 x tile_d2)
// fp32 tile.  Address: ga + 4*(x + y*stride0 + z*stride1).
// LDS result is fully packed: lds[((z*tile_d1)+y)*tile_d0 + x].
// 6-arg builtin form (clang-23 / therock-10.0 lane).
// ------------------------------------------------------------
__device__ __forceinline__ void tdm_load_tile_f32(
    unsigned lds_byte_addr, const void* gptr,
    unsigned tile_d0, unsigned tile_d1, unsigned tile_d2,
    unsigned tens_d0, unsigned tens_d1, unsigned tens_d2,
    unsigned long long stride0, unsigned long long stride1) {
  const unsigned long long ga = (unsigned long long)(size_t)gptr;
  u32x4 g0;
  g0.x = 1u;                                            // count=1, user desc
  g0.y = lds_byte_addr;                                 // lds_addr
  g0.z = (unsigned)(ga & 0xffffffffull);                // global_addr[31:0]
  g0.w = (unsigned)((ga >> 32) & 0x1ffffffull) | (2u << 30);  // type=2
  i32x8 g1;
  g1[0] = (int)(2u << 16);                              // data_size = 4 bytes
  g1[1] = (int)((tens_d0 & 0xffffu) << 16);             // tensor_dim0 lo16
  g1[2] = (int)(((tens_d0 >> 16) & 0xffffu) | ((tens_d1 & 0xffffu) << 16));
  g1[3] = (int)(((tens_d1 >> 16) & 0xffffu) | ((tile_d0 & 0xffffu) << 16));
  g1[4] = (int)((tile_d1 & 0xffffu) | ((tile_d2 & 0xffffu) << 16));
  g1[5] = (int)(stride0 & 0xffffffffull);               // dim0 stride lo32
  g1[6] = (int)(((stride0 >> 32) & 0xffffull) |
                ((stride1 & 0xffffull) << 16));
  g1[7] = (int)((stride1 >> 16) & 0xffffffffull);
  i32x4 g2;
  g2.x = (int)tens_d2;                                  // tensor_dim2
  g2.y = 0; g2.z = 0; g2.w = 0;
  i32x4 g3;
  g3.x = 0; g3.y = 0; g3.z = 0; g3.w = 0;
  i32x8 g4;
  g4[0] = 0; g4[1] = 0; g4[2] = 0; g4[3] = 0;
  g4[4] = 0; g4[5] = 0; g4[6] = 0; g4[7] = 0;
  __builtin_amdgcn_tensor_load_to_lds(g0, g1, g2, g3, g4, 0);
}

// ============================================================
// K1: qkv 1x1 conv (GEMM 1152x384 * 384x36864) + bias + 2x2 maxpool
// grid (9, 576), block 256 (8 waves), dynamic LDS = 96KB.
// TDM pulls the x-tile (32 cols x 2 rows x 384 ch) into LDS once;
// waves consume B fragments from LDS, A (weights) from global/L2.
// 4 accumulators = the 4 full-res phases (dy,dx) of the pool window.
// ============================================================
__global__ __launch_bounds__(256) void k_qkv_pool(
    const float* __restrict__ x, const float* __restrict__ w,
    const float* __restrict__ bias, float* __restrict__ out) {
  extern __shared__ float tb[];  // [384][2][32]
  const int lane = threadIdx.x & 31;
  const int wave = threadIdx.x >> 5;
  const int m_base = blockIdx.x * 128 + wave * 16;
  const int nt = blockIdx.y;              // pooled 16-col tile
  const int hs = nt / 6;
  const int wcol0 = (nt % 6) * 32;        // full-res col start
  const int r = lane & 15;
  const int khalf = (lane >> 4) * 2;      // 0 or 2

  // One TDM descriptor for the whole block's B tile (async, TENSORcnt).
  if (threadIdx.x < 32) {
    const float* gsrc = x + (size_t)(2 * hs) * WF + wcol0;
    tdm_load_tile_f32((unsigned)(size_t)tb, gsrc,
                      /*tile*/ 32u, 2u, 384u,
                      /*tensor*/ 192u, 192u, 384u,
                      /*strides*/ 192ull, (unsigned long long)HWF);
    __builtin_amdgcn_s_wait_tensorcnt((short)0);
  }
  __syncthreads();

  const float* __restrict__ wq = w + (size_t)(m_base + r) * DIMC;

  // per-phase LDS offset for this lane's column: dy*32 + 2*r + dx
  int lofs[4];
#pragma unroll
  for (int p = 0; p < 4; ++p) lofs[p] = (p >> 1) * 32 + 2 * r + (p & 1);

  v8f acc[4];
#pragma unroll
  for (int p = 0; p < 4; ++p) acc[p] = v8zero();

  for (int k0 = 0; k0 < DIMC; k0 += 4) {
    const int kb = k0 + khalf;
    v2f a;
    a.x = wq[kb];
    a.y = wq[kb + 1];
#pragma unroll
    for (int p = 0; p < 4; ++p) {
      v2f b;
      b.x = tb[kb * 64 + lofs[p]];          // K = kb
      b.y = tb[(kb + 1) * 64 + lofs[p]];    // K = kb+1
      acc[p] = wmma4(a, b, acc[p]);
    }
  }

  const int rb = (lane < 16) ? 0 : 8;
  const int ws0 = (nt % 6) * 16;
#pragma unroll
  for (int j = 0; j < 8; ++j) {
    const int m = m_base + j + rb;
    float v = fmaxf(fmaxf(acc[0][j], acc[1][j]), fmaxf(acc[2][j], acc[3][j]));
    out[(size_t)m * HWP + hs * WSP + ws0 + r] = v + bias[m];
  }
}

// ============================================================
// K2: depthwise 3x3 conv (SAME) + bias on [1152, 96, 96]
// ============================================================
__global__ __launch_bounds__(256) void k_dwconv(
    const float* __restrict__ in, const float* __restrict__ wdw,
    const float* __restrict__ bdw, float* __restrict__ out) {
  int idx = blockIdx.x * 256 + threadIdx.x;
  if (idx >= QKVC * HWP) return;
  const int ch = idx / HWP;
  const int rem = idx - ch * HWP;
  const int hs = rem / WSP;
  const int ws = rem - hs * WSP;
  const float* __restrict__ ip = in + (size_t)ch * HWP;
  const float* __restrict__ wp = wdw + ch * 9;
  float s = bdw[ch];
#pragma unroll
  for (int ky = 0; ky < 3; ++ky) {
    const int yy = hs + ky - 1;
    if (yy < 0 || yy >= HSP) continue;
#pragma unroll
    for (int kx = 0; kx < 3; ++kx) {
      const int xx = ws + kx - 1;
      if (xx < 0 || xx >= WSP) continue;
      s = fmaf(wp[ky * 3 + kx], ip[yy * WSP + xx], s);
    }
  }
  out[idx] = s;
}

// ============================================================
// K3: inverse L2 norms of q rows (channels 0..383) and k rows (384..767)
// ============================================================
__global__ __launch_bounds__(256) void k_norms(const float* __restrict__ qk,
                                               float* __restrict__ invn) {
  __shared__ float red[256];
  const int row = blockIdx.x;  // 0..767
  const float* __restrict__ p = qk + (size_t)row * HWP;
  float s = 0.f;
  for (int i = threadIdx.x; i < HWP; i += 256) {
    float v = p[i];
    s = fmaf(v, v, s);
  }
  red[threadIdx.x] = s;
  __syncthreads();
  for (int off = 128; off > 0; off >>= 1) {
    if (threadIdx.x < off) red[threadIdx.x] += red[threadIdx.x + off];
    __syncthreads();
  }
  if (threadIdx.x == 0) invn[row] = 1.0f / fmaxf(sqrtf(red[0]), 1e-12f);
}

// ============================================================
// K4: attn = (q @ k^T) * invnq*invnk*temperature  -> [8,48,48]
// grid 72 (head x 3x3 tiles), block 32 (one wave per 16x16 tile), K=9216
// ============================================================
__global__ __launch_bounds__(32) void k_attn(const float* __restrict__ qk,
                                             const float* __restrict__ invn,
                                             const float* __restrict__ temp,
                                             float* __restrict__ attn) {
  const int bid = blockIdx.x;
  const int head = bid / 9;
  const int t = bid % 9;
  const int Mt = t / 3, Nt = t % 3;
  const int lane = threadIdx.x;
  const int r = lane & 15;
  const int khalf = (lane >> 4) * 2;

  const int qrow = head * CH + Mt * 16 + r;   // A row (c)
  const int krow = head * CH + Nt * 16 + r;   // B col (d)
  const float* __restrict__ qp = qk + (size_t)qrow * HWP;
  const float* __restrict__ kp = qk + (size_t)(DIMC + krow) * HWP;

  v8f acc0 = v8zero(), acc1 = v8zero();
  for (int k0 = 0; k0 < HWP; k0 += 8) {
    int kb = k0 + khalf;
    v2f a0, b0, a1, b1;
    a0.x = qp[kb];     a0.y = qp[kb + 1];
    b0.x = kp[kb];     b0.y = kp[kb + 1];
    a1.x = qp[kb + 4]; a1.y = qp[kb + 5];
    b1.x = kp[kb + 4]; b1.y = kp[kb + 5];
    __builtin_prefetch(qp + kb + 64, 0, 1);
    __builtin_prefetch(kp + kb + 64, 0, 1);
    acc0 = wmma4(a0, b0, acc0);
    acc1 = wmma4(a1, b1, acc1);
  }
  v8f acc = acc0 + acc1;

  const int rb = (lane < 16) ? 0 : 8;
  const int dcol = Nt * 16 + r;
  const float sk = invn[DIMC + head * CH + dcol];
  const float tp = temp[head];
#pragma unroll
  for (int j = 0; j < 8; ++j) {
    const int c = Mt * 16 + j + rb;
    const float sq = invn[head * CH + c];
    attn[(size_t)(head * CH + c) * CH + dcol] = acc[j] * sq * sk * tp;
  }
}

// ============================================================
// K5: combined top-k softmax: P = a1*sm(top24)+a2*sm(top32)
//     +a3*sm(top36)+a4*sm(top38). One thread per attn row.
// ============================================================
__global__ __launch_bounds__(64) void k_topk(
    const float* __restrict__ attn, const float* __restrict__ a1,
    const float* __restrict__ a2, const float* __restrict__ a3,
    const float* __restrict__ a4, float* __restrict__ P) {
  const int head = blockIdx.x;
  const int c = threadIdx.x;
  if (c >= CH) return;
  const float* __restrict__ row = attn + (size_t)(head * CH + c) * CH;

  float av[CH];
  float mx = -3.4e38f;
  for (int j = 0; j < CH; ++j) {
    av[j] = row[j];
    mx = fmaxf(mx, av[j]);
  }
  float e[CH];
  int rank[CH];
  float s24 = 0.f, s32 = 0.f, s36 = 0.f, s38 = 0.f;
  for (int i = 0; i < CH; ++i) {
    const float ai = av[i];
    int rk = 0;
    for (int j = 0; j < CH; ++j)
      rk += (av[j] > ai) || (av[j] == ai && j < i);
    const float ei = __expf(ai - mx);
    if (rk < 24) s24 += ei;
    if (rk < 32) s32 += ei;
    if (rk < 36) s36 += ei;
    if (rk < 38) s38 += ei;
    rank[i] = rk;
    e[i] = ei;
  }
  const float c1 = a1[0] / s24, c2 = a2[0] / s32, c3 = a3[0] / s36,
              c4 = a4[0] / s38;
  float* __restrict__ out = P + (size_t)(head * CH + c) * CH;
  for (int i = 0; i < CH; ++i) {
    float w = 0.f;
    const int rk = rank[i];
    if (rk < 24) w += c1;
    if (rk < 32) w += c2;
    if (rk < 36) w += c3;
    if (rk < 38) w += c4;
    out[i] = e[i] * w;
  }
}

// ============================================================
// K6: g = gelu(P @ v)  -> [384, 9216]
// grid 8*3*576, block 32. A = P[head] (48x48), B = v slab, K=48.
// ============================================================
__global__ __launch_bounds__(32) void k_pv_gelu(const float* __restrict__ P,
                                                const float* __restrict__ qk,
                                                float* __restrict__ g) {
  const int bid = blockIdx.x;
  const int head = bid / (3 * 576);
  const int rem = bid % (3 * 576);
  const int Mt = rem / 576;
  const int nt = rem % 576;
  const int lane = threadIdx.x;
  const int r = lane & 15;
  const int khalf = (lane >> 4) * 2;
  const int n = nt * 16 + r;

  const float* __restrict__ Prow =
      P + (size_t)(head * CH + Mt * 16 + r) * CH;
  const float* __restrict__ vb = qk + (size_t)(2 * DIMC + head * CH) * HWP;

  v8f acc = v8zero();
  for (int k0 = 0; k0 < CH; k0 += 4) {
    const int kb = k0 + khalf;
    v2f a, b;
    a.x = Prow[kb];
    a.y = Prow[kb + 1];
    b.x = vb[(size_t)kb * HWP + n];
    b.y = vb[(size_t)(kb + 1) * HWP + n];
    acc = wmma4(a, b, acc);
  }
  const int rb = (lane < 16) ? 0 : 8;
#pragma unroll
  for (int j = 0; j < 8; ++j) {
    const int c = head * CH + Mt * 16 + j + rb;
    const float xv = acc[j];
    const float gv = 0.5f * xv * (1.0f + erff(xv * 0.70710678118654752f));
    g[(size_t)c * HWP + nt * 16 + r] = gv;
  }
}

// ============================================================
// K7: proj 1x1 conv at pooled res + bias, fused x2 nearest upsample
// grid (3, 576), block 256 (8 waves), dynamic LDS = 24KB.
// TDM pulls the g-tile (16 cols x 384 ch) into LDS once.
// ============================================================
__global__ __launch_bounds__(256) void k_proj_up(
    const float* __restrict__ g, const float* __restrict__ w,
    const float* __restrict__ bias, float* __restrict__ out) {
  extern __shared__ float tg[];  // [384][16]
  const int lane = threadIdx.x & 31;
  const int wave = threadIdx.x >> 5;
  const int m_base = blockIdx.x * 128 + wave * 16;
  const int nt = blockIdx.y;
  const int hs = nt / 6;
  const int ws0 = (nt % 6) * 16;
  const int r = lane & 15;
  const int khalf = (lane >> 4) * 2;

  if (threadIdx.x < 32) {
    const float* gsrc = g + (size_t)nt * 16;
    tdm_load_tile_f32((unsigned)(size_t)tg, gsrc,
                      /*tile*/ 16u, 384u, 0u,
                      /*tensor*/ (unsigned)HWP, 384u, 0u,
                      /*strides*/ (unsigned long long)HWP, 0ull);
    __builtin_amdgcn_s_wait_tensorcnt((short)0);
  }
  __syncthreads();

  const float* __restrict__ wp = w + (size_t)(m_base + r) * DIMC;

  v8f acc = v8zero();
  for (int k0 = 0; k0 < DIMC; k0 += 4) {
    const int kb = k0 + khalf;
    v2f a, b;
    a.x = wp[kb];
    a.y = wp[kb + 1];
    b.x = tg[kb * 16 + r];
    b.y = tg[(kb + 1) * 16 + r];
    acc = wmma4(a, b, acc);
  }

  const int rb = (lane < 16) ? 0 : 8;
  const int ws = ws0 + r;
#pragma unroll
  for (int j = 0; j < 8; ++j) {
    const int m = m_base + j + rb;
    const float v = acc[j] + bias[m];
    float2 vv = make_float2(v, v);
    float* __restrict__ o0 = out + (size_t)m * HWF + (2 * hs) * WF + 2 * ws;
    *reinterpret_cast<float2*>(o0) = vv;
    *reinterpret_cast<float2*>(o0 + WF) = vv;
  }
}

// ============================================================
extern "C" void kernel_launch(void* const* d_in, const int* in_sizes, int n_in,
                              void* d_out, int out_size, void* d_ws,
                              size_t ws_size, hipStream_t stream) {
  const float* x      = (const float*)d_in[0];
  const float* w_qkv  = (const float*)d_in[1];
  const float* b_qkv  = (const float*)d_in[2];
  const float* w_dw   = (const float*)d_in[3];
  const float* b_dw   = (const float*)d_in[4];
  const float* w_proj = (const float*)d_in[5];
  const float* b_proj = (const float*)d_in[6];
  const float* temp   = (const float*)d_in[7];
  const float* a1     = (const float*)d_in[8];
  const float* a2     = (const float*)d_in[9];
  const float* a3     = (const float*)d_in[10];
  const float* a4     = (const float*)d_in[11];
  float* out = (float*)d_out;

  float* wsf = (float*)d_ws;
  float* qkv_pooled = wsf;                                  // 1152*9216
  float* qkv_dw     = qkv_pooled + (size_t)QKVC * HWP;      // 1152*9216
  float* invn       = qkv_dw + (size_t)QKVC * HWP;          // 768
  float* attn       = invn + 2 * DIMC;                      // 8*48*48
  float* P          = attn + HEADS * CH * CH;               // 8*48*48
  float* g          = P + HEADS * CH * CH;                  // 384*9216

  // 1) qkv GEMM + bias + 2x2 maxpool (TDM-fed LDS, 96KB)
  dim3 g1(9, 576);
  k_qkv_pool<<<g1, 256, 384 * 64 * sizeof(float), stream>>>(x, w_qkv, b_qkv,
                                                            qkv_pooled);

  // 2) depthwise 3x3 + bias
  const int n2 = QKVC * HWP;
  k_dwconv<<<(n2 + 255) / 256, 256, 0, stream>>>(qkv_pooled, w_dw, b_dw,
                                                 qkv_dw);

  // 3) q/k inverse L2 norms
  k_norms<<<2 * DIMC, 256, 0, stream>>>(qkv_dw, invn);

  // 4) attn GEMM (WMMA, K=9216) + normalize + temperature
  k_attn<<<72, 32, 0, stream>>>(qkv_dw, invn, temp, attn);

  // 5) combined top-k softmax -> P
  k_topk<<<HEADS, CH, 0, stream>>>(attn, a1, a2, a3, a4, P);

  // 6) P @ v + exact GELU
  k_pv_gelu<<<HEADS * 3 * 576, 32, 0, stream>>>(P, qkv_dw, g);

  // 7) proj GEMM + bias, fused nearest x2 upsample (TDM-fed LDS, 24KB)
  dim3 g7(3, 576);
  k_proj_up<<<g7, 256, 384 * 16 * sizeof(float), stream>>>(g, w_proj, b_proj,
                                                           out);
}